// GPUOptimizedDAG_36764920054219
// MI455X (gfx1250) — compile-verified
//
#include <hip/hip_runtime.h>
#include <math.h>

// Problem constants from the reference
#define DD      32
#define NN      64
#define LAYERS  6
#define S_MAX   416          // DD + LAYERS*NN
#define S_PAD   420          // LDS row stride (bank-conflict free for 16-row column reads)
#define TILE_B  64
#define KB_TOTAL (S_MAX / 4) // 104 K-blocks of 4
#define WLAYER  (KB_TOTAL * NN * 4)   // floats per layer per weight matrix (26624)
#define WMAT    (LAYERS * WLAYER)     // floats per weight matrix (159744)
#define RX_S    68
#define RY_S    36
#define EPSF    1e-6f

typedef float v2f __attribute__((ext_vector_type(2)));
typedef float v8f __attribute__((ext_vector_type(8)));

// ---------------------------------------------------------------------------
// Kernel 1: per-(matrix, layer, n) masked softmax (or one-hot argmax if hard)
// of logits over sources s < valid, written in K-packed layout:
//   wpack[l][s/4][n][s%4]  (so a WMMA B-fragment is one contiguous float2)
// ---------------------------------------------------------------------------
__global__ __launch_bounds__(128)
void route_weights_kernel(const float* __restrict__ lx,
                          const float* __restrict__ ly,
                          float* __restrict__ wpack,
                          const int* __restrict__ hardp) {
    __shared__ float sm[128];
    __shared__ int   si[128];
    __shared__ float ssum[128];

    const int bx    = blockIdx.x;            // 0 .. 2*LAYERS*NN-1
    const int which = bx / (LAYERS * NN);
    const int rem   = bx - which * (LAYERS * NN);
    const int l     = rem >> 6;
    const int n     = rem & 63;
    const int valid = DD + l * NN;
    const int tid   = threadIdx.x;

    const float* row = (which ? ly : lx) + (size_t)(l * NN + n) * S_MAX;
    float* outb = wpack + (size_t)which * WMAT + (size_t)l * WLAYER;

    // max + first-argmax over s < valid
    float m = -3.0e38f; int am = 0x7fffffff;
    for (int s = tid; s < valid; s += 128) {
        float v = row[s];
        if (v > m) { m = v; am = s; }
        else if (v == m && s < am) { am = s; }
    }
    sm[tid] = m; si[tid] = am;
    __syncthreads();
    for (int off = 64; off > 0; off >>= 1) {
        if (tid < off) {
            float vo = sm[tid + off]; int io = si[tid + off];
            if (vo > sm[tid] || (vo == sm[tid] && io < si[tid])) { sm[tid] = vo; si[tid] = io; }
        }
        __syncthreads();
    }
    const float M  = sm[0];
    const int   AM = si[0];

    float acc = 0.f;
    for (int s = tid; s < valid; s += 128) acc += expf(row[s] - M);
    ssum[tid] = acc;
    __syncthreads();
    for (int off = 64; off > 0; off >>= 1) {
        if (tid < off) ssum[tid] += ssum[tid + off];
        __syncthreads();
    }
    const float inv  = 1.0f / ssum[0];
    const int   hard = *hardp;

    for (int s = tid; s < S_MAX; s += 128) {
        float w = 0.f;
        if (s < valid)
            w = hard ? ((s == AM) ? 1.f : 0.f) : expf(row[s] - M) * inv;
        outb[(size_t)(s >> 2) * (NN * 4) + n * 4 + (s & 3)] = w;
    }
}

// ---------------------------------------------------------------------------
// Kernel 2: whole forward pass. One block owns TILE_B batch rows; buf tile
// lives in LDS for all 6 layers. GEMMs use V_WMMA_F32_16X16X4_F32.
// ---------------------------------------------------------------------------
__global__ __launch_bounds__(256)
void dag_forward_kernel(const float* __restrict__ x,
                        const float* __restrict__ wpx,
                        const float* __restrict__ wpy,
                        const float* __restrict__ omega,
                        const float* __restrict__ phi,
                        const float* __restrict__ amp,
                        const float* __restrict__ pexp,
                        const float* __restrict__ alpha,
                        const float* __restrict__ Wout,
                        const float* __restrict__ bout,
                        float* __restrict__ out,
                        int B) {
    __shared__ __align__(16) float bufS[TILE_B * S_PAD];   // 105 KB
    __shared__ __align__(16) float rxs[TILE_B * RX_S];     // rx results (64 cols)
    __shared__ __align__(16) float rys[TILE_B * RY_S];     // ry results (cols 32..63)
    __shared__ float wAri[24 * 4];

    const int tid     = threadIdx.x;
    const int rowBase = blockIdx.x * TILE_B;

    // Load the x tile into buf[:, :32] (columns >= valid are never read before written)
    for (int i = tid; i < TILE_B * DD; i += 256) {
        int r = i >> 5, c = i & 31;
        float v = 0.f;
        if (rowBase + r < B) v = x[(size_t)(rowBase + r) * DD + c];
        bufS[r * S_PAD + c] = v;
    }

    const int lane   = tid & 31;
    const int wave   = tid >> 5;
    const int g      = wave >> 2;         // 0 or 1
    const int mBase  = (wave & 3) * 16;   // M-tile base row
    const int lane15 = lane & 15;
    const int laneHi = lane >> 4;
    const int kOff   = laneHi * 2;        // K sub-offset per WMMA f32 A/B layout

    const int ntA = g ? 2 : 0;            // this wave's rx N-tiles: ntA, ntA+1
    const int ntY = g ? 2 : 3;            // this wave's ry N-tile
    const int nX0 = ntA * 16 + lane15;
    const int nX1 = (ntA + 1) * 16 + lane15;
    const int nY  = ntY * 16 + lane15;
    const int aBase = (mBase + lane15) * S_PAD + kOff;

    __syncthreads();

    for (int l = 0; l < LAYERS; ++l) {
        const int valid = DD + l * NN;

        // per-layer arithmetic-mix softmax weights (tiny)
        if (tid < 24) {
            const float* al = alpha + (size_t)(l * 24 + tid) * 4;
            float a0 = al[0], a1 = al[1], a2 = al[2], a3 = al[3];
            float mx = fmaxf(fmaxf(a0, a1), fmaxf(a2, a3));
            float e0 = expf(a0 - mx), e1 = expf(a1 - mx);
            float e2 = expf(a2 - mx), e3 = expf(a3 - mx);
            float inv = 1.f / (e0 + e1 + e2 + e3);
            wAri[tid * 4 + 0] = e0 * inv;
            wAri[tid * 4 + 1] = e1 * inv;
            wAri[tid * 4 + 2] = e2 * inv;
            wAri[tid * 4 + 3] = e3 * inv;
        }

        const float* wxL = wpx + (size_t)l * WLAYER;
        const float* wyL = wpy + (size_t)l * WLAYER;

        v8f c0 = {}; v8f c1 = {}; v8f c2 = {};
        const int kbEnd = valid >> 2;

        #pragma unroll 2
        for (int kb = 0; kb < kbEnd; ++kb) {
            v2f a   = *(const v2f*)&bufS[aBase + kb * 4];
            v2f bx0 = *(const v2f*)(wxL + (size_t)kb * (NN * 4) + nX0 * 4 + kOff);
            v2f bx1 = *(const v2f*)(wxL + (size_t)kb * (NN * 4) + nX1 * 4 + kOff);
            v2f by  = *(const v2f*)(wyL + (size_t)kb * (NN * 4) + nY  * 4 + kOff);
            c0 = __builtin_amdgcn_wmma_f32_16x16x4_f32(false, a, false, bx0, (short)0, c0, false, false);
            c1 = __builtin_amdgcn_wmma_f32_16x16x4_f32(false, a, false, bx1, (short)0, c1, false, false);
            c2 = __builtin_amdgcn_wmma_f32_16x16x4_f32(false, a, false, by,  (short)0, c2, false, false);
        }

        // D layout: VGPR r holds (M = mBase + laneHi*8 + r, N = tileBase + lane15)
        #pragma unroll
        for (int r = 0; r < 8; ++r) {
            int row = mBase + laneHi * 8 + r;
            rxs[row * RX_S + ntA * 16 + lane15]       = c0[r];
            rxs[row * RX_S + (ntA + 1) * 16 + lane15] = c1[r];
            rys[row * RY_S + (ntY * 16 - 32) + lane15] = c2[r];
        }
        __syncthreads();

        // elementwise epilogue: 64 rows x 64 cols, 16 elements per thread
        #pragma unroll 4
        for (int i = 0; i < 16; ++i) {
            int idx = tid + i * 256;
            int row = idx >> 6;
            int col = idx & 63;
            float vx = rxs[row * RX_S + col];
            float o;
            if (col < 24) {
                o = amp[l * 24 + col] * sinf(omega[l * 24 + col] * vx + phi[l * 24 + col]);
            } else if (col < 40) {
                int j = col - 24;
                float e = pexp[l * 16 + j];
                float sgn = (vx > 0.f) ? 1.f : ((vx < 0.f) ? -1.f : 0.f);
                o = sgn * powf(fabsf(vx) + EPSF, e);
            } else {
                int j = col - 40;
                float xa = vx;
                float ya = rys[row * RY_S + (col - 32)];
                const float* w = &wAri[j * 4];
                float prod = xa * ya;
                o = w[0] * (xa + ya) + w[1] * (xa - ya) + w[2] * prod
                  + w[3] * (prod / (ya * ya + EPSF));
            }
            bufS[row * S_PAD + valid + col] = o;
        }
        __syncthreads();
    }

    // Final matvec: out = buf @ W_out + b_out  (4 partials per row)
    {
        int row  = tid & 63;
        int part = tid >> 6;
        float s = 0.f;
        int cStart = part * 104;
        for (int c = cStart; c < cStart + 104; ++c)
            s += bufS[row * S_PAD + c] * Wout[c];
        rxs[row * 4 + part] = s;
        __syncthreads();
        if (tid < TILE_B && (rowBase + tid) < B) {
            float r = rxs[tid * 4 + 0] + rxs[tid * 4 + 1]
                    + rxs[tid * 4 + 2] + rxs[tid * 4 + 3] + bout[0];
            out[rowBase + tid] = r;
        }
    }
}

extern "C" void kernel_launch(void* const* d_in, const int* in_sizes, int n_in,
                              void* d_out, int out_size, void* d_ws, size_t ws_size,
                              hipStream_t stream) {
    const float* x     = (const float*)d_in[0];
    const float* lx    = (const float*)d_in[1];
    const float* ly    = (const float*)d_in[2];
    const float* omega = (const float*)d_in[3];
    const float* phi   = (const float*)d_in[4];
    const float* amp   = (const float*)d_in[5];
    const float* pexp  = (const float*)d_in[6];
    const float* alpha = (const float*)d_in[7];
    const float* Wout  = (const float*)d_in[8];
    const float* bout  = (const float*)d_in[9];
    const int*   hard  = (const int*)d_in[10];

    const int B = in_sizes[0] / DD;
    float* wpack = (float*)d_ws;   // [2][LAYERS][KB_TOTAL][NN][4] floats (~1.25 MB)

    route_weights_kernel<<<2 * LAYERS * NN, 128, 0, stream>>>(lx, ly, wpack, hard);

    const int nb = (B + TILE_B - 1) / TILE_B;
    dag_forward_kernel<<<nb, 256, 0, stream>>>(x, wpack, wpack + WMAT,
                                               omega, phi, amp, pexp, alpha,
                                               Wout, bout, (float*)d_out, B);
}